// LSTM_33234456936560
// MI455X (gfx1250) — compile-verified
//
#include <hip/hip_runtime.h>

typedef float v2f __attribute__((ext_vector_type(2)));
typedef float v8f __attribute__((ext_vector_type(8)));
typedef unsigned int u32x4 __attribute__((ext_vector_type(4)));
typedef int i32x8 __attribute__((ext_vector_type(8)));
typedef int i32x4 __attribute__((ext_vector_type(4)));

#define BB 32       // batch
#define SS 1024     // seq len
#define II 512      // input dim
#define HH 512      // hidden dim
#define NG 32       // persistent workgroups
#define JPB 16      // hidden units per workgroup
#define NROWS 64    // 4 gates * JPB weight rows per workgroup

// TDM pads 2 DWORDs after every 256 DWORDs stored -> LDS row stride 516 words
// (516 % 64 == 4: B-frag lanes n=0..15 hit banks 4n..4n+3, conflict-free) and
// an extra +2 word shift for columns k >= 256 within a row.
#define RSTR   516
#define TILEW  (JPB * RSTR)            // words per 16-row gate tile  = 8256
#define WREG   (4 * TILEW)             // words per weight region     = 33024

// dynamic LDS layout (floats):
//   Whh[WREG] ++ Wih[WREG] ++ gbuf[4*BB*JPB] ++ c_lds[BB*JPB] ++ bsum[NROWS]
#define LDS_FLOATS (2 * WREG + 4 * BB * JPB + BB * JPB + NROWS)

__device__ __forceinline__ void grid_sync(unsigned* cnt, unsigned target) {
  __threadfence();            // release: my stores visible at device scope
  __syncthreads();            // whole block has fenced
  if (threadIdx.x == 0) {
    atomicAdd(cnt, 1u);
    while (__hip_atomic_load(cnt, __ATOMIC_RELAXED, __HIP_MEMORY_SCOPE_AGENT) < target) {
      __builtin_amdgcn_s_sleep(2);
    }
  }
  __threadfence();            // acquire: drop stale cached h lines
  __syncthreads();
}

#if __has_builtin(__builtin_amdgcn_tensor_load_to_lds)
#define USE_TDM 1
// 2D TDM load: tile_d1 rows x tile_d0 elems (4B), row stride stride0 elems,
// destination LDS with 2-DW pad every 256 DW.
__device__ __forceinline__ void tdm_load_2d(unsigned lds_byte_addr, const void* gptr,
                                            unsigned tile_d0, unsigned tile_d1,
                                            unsigned tensor_d0, unsigned tensor_d1,
                                            unsigned long long stride0) {
  unsigned long long ga = (unsigned long long)(size_t)gptr;
  u32x4 g0;
  g0[0] = 1u;                                              // count=1, user D#
  g0[1] = lds_byte_addr;                                   // lds_addr [63:32]
  g0[2] = (unsigned)(ga & 0xFFFFFFFFu);                    // global_addr lo
  g0[3] = (unsigned)((ga >> 32) & 0x1FFFFFFu) | (2u << 30);// addr[56:32], type=2
  i32x8 g1;
  g1[0] = (int)((2u << 16)      // data_size = 4B
              | (1u << 20)      // pad_enable
              | (7u << 22)      // pad_interval: 256 DWORDs
              | (1u << 25));    // pad_amount:   2 DWORDs
  g1[1] = (int)((tensor_d0 & 0xFFFFu) << 16);              // tensor_dim0[15:0]
  g1[2] = (int)(((tensor_d0 >> 16) & 0xFFFFu) | ((tensor_d1 & 0xFFFFu) << 16));
  g1[3] = (int)(((tensor_d1 >> 16) & 0xFFFFu) | ((tile_d0 & 0xFFFFu) << 16));
  g1[4] = (int)(tile_d1 & 0xFFFFu);                        // tile_dim1, tile_dim2=0
  g1[5] = (int)(unsigned)(stride0 & 0xFFFFFFFFull);        // stride0[31:0]
  g1[6] = (int)(unsigned)((stride0 >> 32) & 0xFFFFull);    // stride0[47:32]
  g1[7] = 0;
  i32x4 z4 = {0, 0, 0, 0};
  i32x8 z8 = {0, 0, 0, 0, 0, 0, 0, 0};
  __builtin_amdgcn_tensor_load_to_lds(g0, g1, z4, z4, z8, 0);
}
#else
#define USE_TDM 0
#endif

__global__ void init_cnt_kernel(unsigned* cnt) { *cnt = 0u; }

__global__ __launch_bounds__(256) void lstm_persistent(
    const float* __restrict__ x,     // [B,S,I]
    const float* __restrict__ W_ih,  // [4H,I]
    const float* __restrict__ W_hh,  // [4H,H]
    const float* __restrict__ b_ih,  // [4H]
    const float* __restrict__ b_hh,  // [4H]
    const float* __restrict__ h0,    // [B,H]
    const float* __restrict__ c0,    // [B,H]
    float* __restrict__ out,         // [B,S,H] ++ h_n[B,H] ++ c_n[B,H]
    float* __restrict__ hbuf0, float* __restrict__ hbuf1,
    unsigned* __restrict__ cnt)
{
  extern __shared__ float smem[];
  float* Whh   = smem;                 // 4 tiles x 16 rows x 516
  float* Wih   = smem + WREG;
  float* gbuf  = smem + 2 * WREG;      // [gate][b][j]
  float* c_lds = gbuf + 4 * BB * JPB;
  float* bsum  = c_lds + BB * JPB;

  const int g      = blockIdx.x;
  const int tid    = threadIdx.x;
  const int lane   = tid & 31;
  const int wv     = tid >> 5;     // 0..7
  const int gate   = wv >> 1;      // 0..3  (N-tile == one gate's 16 j's)
  const int mi     = wv & 1;       // batch half (M-tile)
  const int laneLo = lane & 15;
  const int laneHi = lane >> 4;    // 0/1 -> K sub-pair per WMMA f32 layout

  // ---- preamble: stage weight slice into LDS ----
#if USE_TDM
  {
    // one TDM descriptor per wave: wave (gate, src) loads 16 rows x 512 f32
    const float* src  = (wv & 1) ? W_ih : W_hh;
    float*       dstp = ((wv & 1) ? Wih : Whh) + gate * TILEW;
    unsigned lds_addr = (unsigned)(size_t)(void*)dstp;           // flat->LDS offset
    const float* gptr = src + (size_t)(gate * HH + g * JPB) * 512;
    tdm_load_2d(lds_addr, gptr, /*tile*/ 512, JPB, /*tensor*/ 512, 4 * HH, 512ull);
    __builtin_amdgcn_s_wait_tensorcnt(0);
  }
#else
  for (int idx = tid; idx < NROWS * 128; idx += 256) {
    int r  = idx >> 7;           // 0..63: gate t = r>>4, local j = r&15
    int c4 = idx & 127;          // float4 column (0..127) within 512-col row
    int t  = r >> 4, jl = r & 15;
    size_t grow = (size_t)(t * HH + g * JPB + jl);
    int k = c4 * 4;
    int ofs = (t * JPB + jl) * RSTR + k + 2 * (k >= 256 ? 1 : 0);
    *(float4*)(Whh + ofs) = *(const float4*)(W_hh + grow * HH + k);
    *(float4*)(Wih + ofs) = *(const float4*)(W_ih + grow * II + k);
  }
#endif
  if (tid < NROWS) {
    int tt = tid >> 4, jl = tid & 15;
    int col = tt * HH + g * JPB + jl;
    bsum[tid] = b_ih[col] + b_hh[col];
  }
  for (int v = tid; v < BB * JPB; v += 256) {
    int b = v >> 4, j = v & 15;
    c_lds[v] = c0[b * HH + g * JPB + j];
    hbuf0[b * HH + g * JPB + j] = h0[b * HH + g * JPB + j];
  }
  __syncthreads();
  unsigned phase = 1;
  grid_sync(cnt, phase * NG);     // hbuf0 initialized device-wide

  // ---- per-wave constant pointers ----
  // B-frag from LDS: lane holds local row n = laneLo; k = 4*kk + 2*laneHi + {0,1};
  // +2 word shift for k >= 256 (TDM pad gap)
  const float* bh = Whh + gate * TILEW + laneLo * RSTR + 2 * laneHi;
  const float* bx = Wih + gate * TILEW + laneLo * RSTR + 2 * laneHi;
  // A-frag from global: lane holds batch row m, elements k = 4*kk + 2*laneHi + {0,1}
  const int    m       = mi * 16 + laneLo;
  const float* aptr_xb = x + (size_t)m * SS * II + 2 * laneHi;

  for (int s = 0; s < SS; ++s) {
    const float* hcur = (s & 1) ? hbuf1 : hbuf0;
    float*       hnxt = (s & 1) ? hbuf0 : hbuf1;
    const float* aptr_h = hcur + m * HH + 2 * laneHi;
    const float* aptr_x = aptr_xb + (size_t)s * II;

    // prefetch next timestep's x slice (8 wave-slots x 64-word offsets per row)
    if (s + 1 < SS) {
      const float* pf = x + (size_t)m * SS * II + (size_t)(s + 1) * II
                          + (gate * 2 + laneHi) * 64;
      __builtin_prefetch(pf, 0, 2);
    }

    v8f acc = {0.f, 0.f, 0.f, 0.f, 0.f, 0.f, 0.f, 0.f};

    // recurrent part: K = 0..511 over h (split at k=256 for the TDM pad gap)
#pragma unroll 8
    for (int kk = 0; kk < 64; ++kk) {
      v2f a = *(const v2f*)(aptr_h + 4 * kk);
      v2f b = *(const v2f*)(bh + 4 * kk);
      acc = __builtin_amdgcn_wmma_f32_16x16x4_f32(false, a, false, b,
                                                  (short)0, acc, false, false);
    }
#pragma unroll 8
    for (int kk = 64; kk < 128; ++kk) {
      v2f a = *(const v2f*)(aptr_h + 4 * kk);
      v2f b = *(const v2f*)(bh + 4 * kk + 2);
      acc = __builtin_amdgcn_wmma_f32_16x16x4_f32(false, a, false, b,
                                                  (short)0, acc, false, false);
    }
    // input part: K over x[:, s, :]
#pragma unroll 8
    for (int kk = 0; kk < 64; ++kk) {
      v2f a = *(const v2f*)(aptr_x + 4 * kk);
      v2f b = *(const v2f*)(bx + 4 * kk);
      acc = __builtin_amdgcn_wmma_f32_16x16x4_f32(false, a, false, b,
                                                  (short)0, acc, false, false);
    }
#pragma unroll 8
    for (int kk = 64; kk < 128; ++kk) {
      v2f a = *(const v2f*)(aptr_x + 4 * kk);
      v2f b = *(const v2f*)(bx + 4 * kk + 2);
      acc = __builtin_amdgcn_wmma_f32_16x16x4_f32(false, a, false, b,
                                                  (short)0, acc, false, false);
    }

    // scatter pre-activations: acc[v] @ lane -> M = v + 8*laneHi, N = laneLo
#pragma unroll
    for (int v = 0; v < 8; ++v) {
      int mloc = mi * 16 + v + 8 * laneHi;
      gbuf[gate * (BB * JPB) + mloc * JPB + laneLo] = acc[v];
    }
    __syncthreads();

    // elementwise gate update: 512 cells, 2 per thread
#pragma unroll
    for (int u = 0; u < 2; ++u) {
      int v = tid + u * 256;
      int b = v >> 4, j = v & 15;
      float pi = gbuf[0 * 512 + v] + bsum[0  + j];
      float pf = gbuf[1 * 512 + v] + bsum[16 + j];
      float pg = gbuf[2 * 512 + v] + bsum[32 + j];
      float po = gbuf[3 * 512 + v] + bsum[48 + j];
      float ig = 1.f / (1.f + __expf(-pi));
      float fg = 1.f / (1.f + __expf(-pf));
      float gg = tanhf(pg);
      float og = 1.f / (1.f + __expf(-po));
      float cn = fg * c_lds[v] + ig * gg;
      float hn = og * tanhf(cn);
      c_lds[v] = cn;
      int col = g * JPB + j;
      __builtin_nontemporal_store(hn, out + ((size_t)b * SS + s) * HH + col);
      hnxt[b * HH + col] = hn;
      if (s == SS - 1) {
        out[(size_t)BB * SS * HH + (size_t)b * HH + col] = hn;                   // h_n
        out[(size_t)BB * SS * HH + (size_t)BB * HH + (size_t)b * HH + col] = cn; // c_n
      }
    }
    ++phase;
    grid_sync(cnt, phase * NG);   // h(s+1) visible everywhere before next read
  }
}

extern "C" void kernel_launch(void* const* d_in, const int* in_sizes, int n_in,
                              void* d_out, int out_size, void* d_ws, size_t ws_size,
                              hipStream_t stream) {
  const float* x    = (const float*)d_in[0];
  const float* W_ih = (const float*)d_in[1];
  const float* W_hh = (const float*)d_in[2];
  const float* b_ih = (const float*)d_in[3];
  const float* b_hh = (const float*)d_in[4];
  const float* h0   = (const float*)d_in[5];
  const float* c0   = (const float*)d_in[6];
  float* out = (float*)d_out;

  // workspace layout: [0..255] barrier counter, then two h ping-pong buffers
  unsigned* cnt = (unsigned*)d_ws;
  float* hbuf0 = (float*)((char*)d_ws + 256);
  float* hbuf1 = hbuf0 + BB * HH;

  init_cnt_kernel<<<1, 1, 0, stream>>>(cnt);
  lstm_persistent<<<NG, 256, LDS_FLOATS * sizeof(float), stream>>>(
      x, W_ih, W_hh, b_ih, b_hh, h0, c0, out, hbuf0, hbuf1, cnt);
}